// PointPillarsVoxelization_44057774522682
// MI455X (gfx1250) — compile-verified
//
#include <hip/hip_runtime.h>
#include <stdint.h>
#include <stddef.h>

// ---------------- compile-time grid constants (from reference) ----------------
#define GX1      433            // int((69.12-0)/0.16) + 1
#define GY1      497            // int((39.68+39.68)/0.16) + 1
#define GZ1      2              // int((1+3)/4) + 1
#define NCELL    (GX1*GY1*GZ1)  // 430402
#define SENTINEL ((unsigned)NCELL)
#define MAX_PTS  32
#define MAX_VOX  40000
#define DIMS     9
#define GRID_X   432            // GRID[0]
#define GRID_Y   496            // GRID[1]

#define SCAN_TILE 1024
#define NBLK_SCAN ((NCELL + SCAN_TILE - 1) / SCAN_TILE)   // 421

static_assert(NBLK_SCAN <= 512, "single-block top-level scan assumption");

#define VOXF   ((size_t)MAX_VOX * MAX_PTS * DIMS)   // 11,520,000 floats (zeros)
#define COORDF ((size_t)MAX_VOX * 3)                // 120,000 floats (-1 default)
#define NUMF   ((size_t)MAX_VOX)                    // 40,000 floats (0 default)

__device__ __forceinline__ int clampi(int v, int lo, int hi) {
  return v < lo ? lo : (v > hi ? hi : v);
}

// ---------------- K0: init coords region to -1.0f with b128 stores -----------
// (voxels + num_points regions are zero-filled by hipMemsetAsync on the host
//  side; only coords needs the non-byte-replicable -1.0f pattern)
__global__ void k_init_coords(float* __restrict__ coords) {
  // COORDF == 120,000, divisible by 4 -> 30,000 float4 stores
  const int n4 = (int)(COORDF / 4);
  int i = blockIdx.x * blockDim.x + threadIdx.x;
  if (i < n4) {
    float4 v = make_float4(-1.0f, -1.0f, -1.0f, -1.0f);
    ((float4*)coords)[i] = v;
  }
}

// ---------------- K1: per-point voxel linear id + histogram ------------------
__global__ void k_points(const float* __restrict__ pf, int N,
                         unsigned* __restrict__ lin,
                         unsigned* __restrict__ counts) {
  int p = blockIdx.x * blockDim.x + threadIdx.x;
  if (p >= N) return;
  const float x = pf[(size_t)p * DIMS + 0];
  const float y = pf[(size_t)p * DIMS + 1];
  const float z = pf[(size_t)p * DIMS + 2];
  const bool in_range =
      (x >= 0.0f)    && (x < 69.12f) &&
      (y >= -39.68f) && (y < 39.68f) &&
      (z >= -3.0f)   && (z < 1.0f);
  int vx = clampi((int)floorf((x - 0.0f)   / 0.16f), 0, GX1 - 1);
  int vy = clampi((int)floorf((y + 39.68f) / 0.16f), 0, GY1 - 1);
  int vz = clampi((int)floorf((z + 3.0f)   / 4.0f),  0, GZ1 - 1);
  unsigned l = in_range ? (unsigned)((vz * GY1 + vy) * GX1 + vx) : SENTINEL;
  lin[p] = l;
  if (l < SENTINEL) atomicAdd(&counts[l], 1u);
}

// ---------------- K2a: per-tile exclusive scans (counts & occupancy) ---------
__global__ void k_scan_partial(const unsigned* __restrict__ counts,
                               unsigned* __restrict__ offsets,
                               unsigned* __restrict__ voxid,
                               unsigned* __restrict__ bsumC,
                               unsigned* __restrict__ bsumO) {
  __shared__ unsigned sc[256], so[256];
  const int t = threadIdx.x;
  const int base = blockIdx.x * SCAN_TILE + t * 4;
  unsigned c[4], o[4], sumc = 0, sumo = 0;
#pragma unroll
  for (int i = 0; i < 4; ++i) {
    int idx = base + i;
    c[i] = (idx < NCELL) ? counts[idx] : 0u;
    o[i] = (c[i] > 0u) ? 1u : 0u;
    sumc += c[i]; sumo += o[i];
  }
  sc[t] = sumc; so[t] = sumo;
  __syncthreads();
  for (int d = 1; d < 256; d <<= 1) {
    unsigned a = 0, b = 0;
    if (t >= d) { a = sc[t - d]; b = so[t - d]; }
    __syncthreads();
    sc[t] += a; so[t] += b;
    __syncthreads();
  }
  unsigned rc = sc[t] - sumc;   // exclusive within block
  unsigned ro = so[t] - sumo;
#pragma unroll
  for (int i = 0; i < 4; ++i) {
    int idx = base + i;
    if (idx < NCELL) { offsets[idx] = rc; voxid[idx] = ro; }
    rc += c[i]; ro += o[i];
  }
  if (t == 255) { bsumC[blockIdx.x] = sc[t]; bsumO[blockIdx.x] = so[t]; }
}

// ---------------- K2b: scan the tile sums (single block) ---------------------
__global__ void k_scan_block(unsigned* __restrict__ bsumC,
                             unsigned* __restrict__ bsumO) {
  __shared__ unsigned sc[512], so[512];
  const int t = threadIdx.x;
  unsigned c = (t < NBLK_SCAN) ? bsumC[t] : 0u;
  unsigned o = (t < NBLK_SCAN) ? bsumO[t] : 0u;
  sc[t] = c; so[t] = o;
  __syncthreads();
  for (int d = 1; d < 512; d <<= 1) {
    unsigned a = 0, b = 0;
    if (t >= d) { a = sc[t - d]; b = so[t - d]; }
    __syncthreads();
    sc[t] += a; so[t] += b;
    __syncthreads();
  }
  if (t < NBLK_SCAN) { bsumC[t] = sc[t] - c; bsumO[t] = so[t] - o; }
}

// ---------------- K2c: add tile offsets --------------------------------------
__global__ void k_scan_add(unsigned* __restrict__ offsets,
                           unsigned* __restrict__ voxid,
                           const unsigned* __restrict__ bsumC,
                           const unsigned* __restrict__ bsumO) {
  int i = blockIdx.x * blockDim.x + threadIdx.x;
  if (i < NCELL) {
    int b = i / SCAN_TILE;
    offsets[i] += bsumC[b];
    voxid[i]   += bsumO[b];
  }
}

// ---------------- K3: scatter point indices into per-cell segments ----------
__global__ void k_scatter(const unsigned* __restrict__ lin, int N,
                          const unsigned* __restrict__ offsets,
                          unsigned* __restrict__ cursor,
                          unsigned* __restrict__ plist) {
  int p = blockIdx.x * blockDim.x + threadIdx.x;
  if (p >= N) return;
  unsigned c = lin[p];
  if (c < SENTINEL) {
    unsigned slot = atomicAdd(&cursor[c], 1u);
    plist[offsets[c] + slot] = (unsigned)p;
  }
}

// ---------------- K4: one wave32 per cell: rank-by-count + gather ------------
// rank(e) = #{j in cell : idx_j < idx_e}  == stable-sort rank (indices unique),
// which matches the reference's stable argsort within a voxel exactly.
__global__ void k_emit(const float* __restrict__ pf,
                       const unsigned* __restrict__ counts,
                       const unsigned* __restrict__ offsets,
                       const unsigned* __restrict__ voxid,
                       const unsigned* __restrict__ plist,
                       float* __restrict__ out) {
  float* coords = out + VOXF;
  float* nump   = coords + COORDF;

  const int wave = (int)((blockIdx.x * blockDim.x + threadIdx.x) >> 5);
  const int lane = (int)(threadIdx.x & 31);
  if (wave >= NCELL) return;
  const int cell = wave;

  const unsigned cnt = counts[cell];
  if (cnt == 0u) return;                 // uniform per wave
  const unsigned v = voxid[cell];
  if (v >= MAX_VOX) return;

  const int vz  = cell / (GX1 * GY1);
  const int rem = cell % (GX1 * GY1);
  const int vy  = rem / GX1;
  const int vx  = rem % GX1;
  if (!((vy < GRID_Y) && (vx < GRID_X))) return;  // in_b filter

  const unsigned off = offsets[cell];
  const unsigned m   = (cnt < (unsigned)MAX_PTS) ? cnt : (unsigned)MAX_PTS;

  if (lane == 0) {
    coords[(size_t)v * 3 + 0] = (float)vz;
    coords[(size_t)v * 3 + 1] = (float)vy;
    coords[(size_t)v * 3 + 2] = (float)vx;
    nump[v] = (float)m;
  }

  for (unsigned base = 0; base < cnt; base += 32u) {
    const unsigned gi = base + (unsigned)lane;
    const int e = (gi < cnt) ? (int)plist[off + gi] : 0x7fffffff;
    if (gi < cnt) {
      // gfx1250 global_prefetch_b8: warm the gathered feature row
      __builtin_prefetch(pf + (size_t)e * DIMS, 0, 0);
    }
    int rank = 0;
    for (unsigned jb = 0; jb < cnt; jb += 32u) {
      const unsigned jj = jb + (unsigned)lane;
      const int pj = (jj < cnt) ? (int)plist[off + jj] : 0x7fffffff;
#pragma unroll
      for (int k = 0; k < 32; ++k) {
        const int o = __shfl(pj, k, 32);   // wave32 cross-lane (ds_bpermute)
        rank += (o < e) ? 1 : 0;
      }
    }
    if (gi < cnt && rank < MAX_PTS) {
      const float* __restrict__ src = pf + (size_t)e * DIMS;
      float* __restrict__ dst = out + ((size_t)v * MAX_PTS + (size_t)rank) * DIMS;
#pragma unroll
      for (int d = 0; d < DIMS; ++d) dst[d] = src[d];
    }
  }
}

// ---------------- host-side launcher -----------------------------------------
extern "C" void kernel_launch(void* const* d_in, const int* in_sizes, int n_in,
                              void* d_out, int out_size, void* d_ws, size_t ws_size,
                              hipStream_t stream) {
  (void)n_in; (void)out_size; (void)ws_size;
  const float* pf = (const float*)d_in[0];
  float* out = (float*)d_out;
  const int N = in_sizes[0] / DIMS;   // points

  // workspace layout (all u32); counts & cursor adjacent for one memset
  unsigned* lin     = (unsigned*)d_ws;
  unsigned* plist   = lin + N;
  unsigned* counts  = plist + N;
  unsigned* cursor  = counts + NCELL;
  unsigned* offsets = cursor + NCELL;
  unsigned* voxid   = offsets + NCELL;
  unsigned* bsumC   = voxid + NCELL;
  unsigned* bsumO   = bsumC + NBLK_SCAN;

  const int TB = 256;
  const int nCellBlk = (NCELL + TB - 1) / TB;
  const int nPtBlk   = (N + TB - 1) / TB;

  // Zero-fill: voxel features (11.52M floats) + num_points tail via DMA memset.
  // Layout in d_out: [voxels | coords | num_points]; voxels and num_points are
  // both all-zero bytes, coords needs -1.0f (kernel below).
  hipMemsetAsync(out, 0, VOXF * sizeof(float), stream);
  hipMemsetAsync(out + VOXF + COORDF, 0, NUMF * sizeof(float), stream);
  hipMemsetAsync(counts, 0, (size_t)2 * NCELL * sizeof(unsigned), stream);

  k_init_coords<<<(int)((COORDF / 4 + TB - 1) / TB), TB, 0, stream>>>(out + VOXF);
  k_points     <<<nPtBlk, TB, 0, stream>>>(pf, N, lin, counts);
  k_scan_partial<<<NBLK_SCAN, TB, 0, stream>>>(counts, offsets, voxid, bsumC, bsumO);
  k_scan_block <<<1, 512, 0, stream>>>(bsumC, bsumO);
  k_scan_add   <<<nCellBlk, TB, 0, stream>>>(offsets, voxid, bsumC, bsumO);
  k_scatter    <<<nPtBlk, TB, 0, stream>>>(lin, N, offsets, cursor, plist);
  k_emit       <<<(NCELL + 7) / 8, TB, 0, stream>>>(pf, counts, offsets, voxid, plist, out);
}